// FlowMatching_44409961840992
// MI455X (gfx1250) — compile-verified
//
#include <hip/hip_runtime.h>

typedef __attribute__((ext_vector_type(2))) float v2f;
typedef __attribute__((ext_vector_type(8))) float v8f;

#define TILE 16

// Branchless ordered top-3 (smallest) insertion.
__device__ __forceinline__ void top3_insert(float v, int idx,
                                            float& d0, float& d1, float& d2,
                                            int& i0, int& i1, int& i2) {
  bool lt0 = v < d0;
  bool lt1 = v < d1;
  bool lt2 = v < d2;
  d2 = lt1 ? d1 : (lt2 ? v : d2);
  i2 = lt1 ? i1 : (lt2 ? idx : i2);
  d1 = lt0 ? d0 : (lt1 ? v : d1);
  i1 = lt0 ? i0 : (lt1 ? idx : i1);
  d0 = lt0 ? v  : d0;
  i0 = lt0 ? idx : i0;
}

// Pack per-atom A-matrix operand data:
//   P0[i] = (x, y)                     -> consumed by lanes 0..15  (K=0,1)
//   P1[i] = (z, ctx ? ||x||^2 : +INF)  -> consumed by lanes 16..31 (K=2,3)
// Poisoning ||x||^2 with +INF makes every non-context candidate's distance
// +INF straight out of the WMMA accumulator: masking is free in the hot loop.
__global__ __launch_bounds__(256) void chem_prior_prep(
    const float* __restrict__ X,
    const int* __restrict__ block_ids,
    const unsigned char* __restrict__ gmask,
    float2* __restrict__ P0,
    float2* __restrict__ P1,
    int n)
{
  int i = blockIdx.x * blockDim.x + threadIdx.x;
  if (i >= n) return;
  float x = X[3 * i + 0];
  float y = X[3 * i + 1];
  float z = X[3 * i + 2];
  float xn = __builtin_fmaf(x, x, __builtin_fmaf(y, y, z * z));
  bool gen = gmask[block_ids[i]] != 0;   // generated atom -> NOT a context atom
  P0[i] = make_float2(x, y);
  P1[i] = make_float2(z, gen ? __builtin_inff() : xn);
}

// One wave32 per 16 query atoms. For each 16-candidate tile:
//   D[m][n] = -2 * x_c[m] . x_q[n] + ||x_c[m]||^2     via v_wmma_f32_16x16x4_f32
// (adding the lane-constant ||x_q||^2 later preserves top-3 ordering).
// Software-pipelined one tile ahead; insert chain gated by a wave-uniform
// "any lane improved" test so the common-case loop body is ~a dozen ops.
__global__ __launch_bounds__(256) void chem_prior_topk(
    const float* __restrict__ X,
    const float2* __restrict__ P0,
    const float2* __restrict__ P1,
    const float* __restrict__ base_noise,
    const float* __restrict__ centroid_noise,
    const int* __restrict__ block_ids,
    const unsigned char* __restrict__ gmask,
    float* __restrict__ out,
    int n)
{
  const int lane    = threadIdx.x & 31;
  const int wave    = (blockIdx.x * blockDim.x + threadIdx.x) >> 5;
  const int qbase   = wave * TILE;
  const int hi      = lane >> 4;     // 0: K=0,1 half; 1: K=2,3 half
  const int col     = lane & 15;     // query column this lane owns in C/D
  const int q       = qbase + col;

  // Query data (loop-invariant).
  float qx = X[3 * q + 0];
  float qy = X[3 * q + 1];
  float qz = X[3 * q + 2];
  float qsq = __builtin_fmaf(qx, qx, __builtin_fmaf(qy, qy, qz * qz));

  // B-matrix operand: rows (-2x, -2y, -2z, 1) for this lane's K pair.
  v2f bv;
  bv.x = hi ? (-2.0f * qz) : (-2.0f * qx);
  bv.y = hi ? 1.0f         : (-2.0f * qy);

  // Per-half A-operand stream (one aligned b64 load per lane per tile).
  const float2* __restrict__ Psel = hi ? P1 : P0;

  float d0 = __builtin_inff(), d1 = __builtin_inff(), d2 = __builtin_inff();
  int   i0 = 0, i1 = 0, i2 = 0;

  const int ntiles = n / TILE;

  // Prologue: load tile 0's A operand.
  float2 a0 = Psel[col];
  v2f av; av.x = a0.x; av.y = a0.y;

  for (int t = 0; t < ntiles; ++t) {
    // Prefetch next tile's A operand (wraps harmlessly on the last iter) so
    // the load is in flight while we do the WMMA + filtering below.
    const int tn = (t + 1 < ntiles) ? (t + 1) : 0;
    float2 anext = Psel[tn * TILE + col];

    v8f acc = {};
    acc = __builtin_amdgcn_wmma_f32_16x16x4_f32(
        /*neg_a=*/false, av, /*neg_b=*/false, bv,
        /*c_mod=*/(short)0, acc, /*reuse_a=*/false, /*reuse_b=*/false);

    // Cheap reject: only run the insert chain if some lane can improve.
    float m01 = fminf(acc[0], acc[1]);
    float m23 = fminf(acc[2], acc[3]);
    float m45 = fminf(acc[4], acc[5]);
    float m67 = fminf(acc[6], acc[7]);
    float m   = fminf(fminf(m01, m23), fminf(m45, m67));

    if (__any(m < d2)) {
      // Lane holds rows M = (hi?8:0)..(hi?15:7) of this candidate tile.
      const int mbase = t * TILE + (hi << 3);
#pragma unroll
      for (int r = 0; r < 8; ++r) {
        top3_insert(acc[r], mbase + r, d0, d1, d2, i0, i1, i2);
      }
    }

    av.x = anext.x; av.y = anext.y;
  }

  // Merge the two half-wave partial top-3 lists for the same query column.
  float e0 = __shfl_xor(d0, 16, 32);
  float e1 = __shfl_xor(d1, 16, 32);
  float e2 = __shfl_xor(d2, 16, 32);
  int   j0 = __shfl_xor(i0, 16, 32);
  int   j1 = __shfl_xor(i1, 16, 32);
  int   j2 = __shfl_xor(i2, 16, 32);
  top3_insert(e0, j0, d0, d1, d2, i0, i1, i2);
  top3_insert(e1, j1, d0, d1, d2, i0, i1, i2);
  top3_insert(e2, j2, d0, d1, d2, i0, i1, i2);

  if (hi == 0) {
    // Reconstruct true distances: stored value + ||x_q||^2 == squared distance.
    float s0 = fmaxf(d0 + qsq, 0.0f) + 1e-12f;
    float s1 = fmaxf(d1 + qsq, 0.0f) + 1e-12f;
    float s2 = fmaxf(d2 + qsq, 0.0f) + 1e-12f;
    float r0 = __builtin_sqrtf(s0);
    float r1 = __builtin_sqrtf(s1);
    float r2 = __builtin_sqrtf(s2);
    float w0 = 1.0f / (r0 + 1e-8f);
    float w1 = 1.0f / (r1 + 1e-8f);
    float w2 = 1.0f / (r2 + 1e-8f);
    float winv = 1.0f / (w0 + w1 + w2);
    w0 *= winv; w1 *= winv; w2 *= winv;

    float cx = w0 * X[3 * i0 + 0] + w1 * X[3 * i1 + 0] + w2 * X[3 * i2 + 0];
    float cy = w0 * X[3 * i0 + 1] + w1 * X[3 * i1 + 1] + w2 * X[3 * i2 + 1];
    float cz = w0 * X[3 * i0 + 2] + w1 * X[3 * i1 + 2] + w2 * X[3 * i2 + 2];

    bool gen = gmask[block_ids[q]] != 0;
    float ox = gen ? __builtin_fmaf(0.5f, centroid_noise[3 * q + 0], cx) : base_noise[3 * q + 0];
    float oy = gen ? __builtin_fmaf(0.5f, centroid_noise[3 * q + 1], cy) : base_noise[3 * q + 1];
    float oz = gen ? __builtin_fmaf(0.5f, centroid_noise[3 * q + 2], cz) : base_noise[3 * q + 2];
    out[3 * q + 0] = ox;
    out[3 * q + 1] = oy;
    out[3 * q + 2] = oz;
  }
}

extern "C" void kernel_launch(void* const* d_in, const int* in_sizes, int n_in,
                              void* d_out, int out_size, void* d_ws, size_t ws_size,
                              hipStream_t stream) {
  const float*         X              = (const float*)d_in[0];
  const float*         base_noise     = (const float*)d_in[1];
  const float*         centroid_noise = (const float*)d_in[2];
  const int*           block_ids      = (const int*)d_in[3];
  const unsigned char* gmask          = (const unsigned char*)d_in[4];  // jnp bool = 1 byte
  float* out = (float*)d_out;

  const int n = in_sizes[0] / 3;           // 16384 atoms

  float2* P0 = (float2*)d_ws;              // n * 8 bytes
  float2* P1 = P0 + n;                     // n * 8 bytes

  chem_prior_prep<<<(n + 255) / 256, 256, 0, stream>>>(X, block_ids, gmask, P0, P1, n);

  // One wave per 16 queries; 8 waves (256 threads) per block.
  const int waves  = n / TILE;             // 1024
  const int blocks = waves / 8;            // 128
  chem_prior_topk<<<blocks, 256, 0, stream>>>(X, P0, P1, base_noise, centroid_noise,
                                              block_ids, gmask, out, n);
}